// BaseCompressor2_57062935495412
// MI455X (gfx1250) — compile-verified
//
#include <hip/hip_runtime.h>
#include <math.h>

#define N_BATCH 32
#define LEN     262144
#define IIRLEN  16384
#define CHUNK   4096
#define NBLK    (LEN / CHUNK)     // 64 blocks per row
#define WAVES   8
#define WSAMP   512               // samples per wave
#define TILE    256               // samples per WMMA tile
#define NTILE   (WSAMP / TILE)    // 2

typedef float v2f __attribute__((ext_vector_type(2)));
typedef float v8f __attribute__((ext_vector_type(8)));

__device__ __forceinline__ float sigmoidf_(float x) { return 1.0f / (1.0f + expf(-x)); }

// -------- Pass 1: block-local truncated-IIR scan via WMMA triangular matmul --------
__global__ __launch_bounds__(256) void pass1_scan(
    const float* __restrict__ x, const float* __restrict__ z_alpha,
    float* __restrict__ w_ws, float* __restrict__ P_ws)
{
  __shared__ float b_lds[CHUNK];
  __shared__ float zend[WAVES * 16];
  __shared__ float Wend[WAVES];

  const int n   = blockIdx.y;
  const int j   = blockIdx.x;
  const int t0  = j * CHUNK;
  const int tid = threadIdx.x;
  const int wv  = tid >> 5;
  const int lane = tid & 31;
  const int m   = lane & 15;   // M-row (A) / N-col (B,C,D) index
  const int hi  = lane >> 4;   // half-wave select

  const float a     = sigmoidf_(z_alpha[n]);
  const float l2a   = log2f(a);
  const float omA   = 1.0f - a;
  const float aK    = exp2f((float)IIRLEN * l2a);   // == 0 in fp32 for realistic alpha

  const float* x0 = x + (size_t)n * (2 * LEN);
  const float* x1 = x0 + LEN;

  // Stage b[t] = (1-a)*(e[t] - a^K * e[t-K]) into LDS, coalesced.
  #pragma unroll
  for (int i = 0; i < CHUNK / 256; ++i) {
    int tl = i * 256 + tid;
    int t  = t0 + tl;
    float v0 = x0[t], v1 = x1[t];
    float e  = 0.5f * (v0 * v0 + v1 * v1);
    float et = 0.0f;
    if (aK > 0.0f) {                 // block-uniform branch; never taken in practice
      int t2 = t - IIRLEN;
      if (t2 >= 0) { float u0 = x0[t2], u1 = x1[t2]; et = 0.5f * (u0 * u0 + u1 * u1); }
    }
    b_lds[tl] = omA * (e - aK * et);
  }
  __syncthreads();

  // B-operand: upper-triangular U[k][i] = a^(i-k) (i>=k), sliced K=4 per WMMA.
  // Layout: VGPR0 lanes0-15 row 4c+0 / lanes16-31 row 4c+2; VGPR1 rows 4c+1 / 4c+3.
  v2f U[4];
  #pragma unroll
  for (int c = 0; c < 4; ++c) {
    int r0 = 4 * c + 2 * hi;
    int r1 = r0 + 1;
    float u0 = (m >= r0) ? exp2f((float)(m - r0) * l2a) : 0.0f;
    float u1 = (m >= r1) ? exp2f((float)(m - r1) * l2a) : 0.0f;
    U[c] = (v2f){u0, u1};
  }

  const float a16  = exp2f(16.0f * l2a);
  const float api  = exp2f((float)(m + 1) * l2a);
  const float a256 = exp2f(256.0f * l2a);
  const float a512 = exp2f(512.0f * l2a);
  float a16p[8];
  #pragma unroll
  for (int v = 0; v < 8; ++v) a16p[v] = exp2f(16.0f * (float)(v + 8 * hi) * l2a);

  float yreg[NTILE][8];
  float carry_tile = 0.0f;   // y just before current tile (wave-local scan)

  #pragma unroll
  for (int tau = 0; tau < NTILE; ++tau) {
    // A-operand: lane m holds segment m, K elements {4c+2hi, 4c+2hi+1}.
    const int abase = wv * WSAMP + tau * TILE + m * 16 + 2 * hi;
    v8f acc = {0.f, 0.f, 0.f, 0.f, 0.f, 0.f, 0.f, 0.f};
    #pragma unroll
    for (int c = 0; c < 4; ++c) {
      const float* p = &b_lds[abase + 4 * c];
      v2f A = (v2f){p[0], p[1]};
      acc = __builtin_amdgcn_wmma_f32_16x16x4_f32(
          false, A, false, U[c], (short)0, acc, false, false);
    }
    __syncthreads();   // previous iteration's zend reads complete before rewrite
    // Segment-end values Z[seg][15] live at N==15: lane15 (segs 0-7), lane31 (segs 8-15)
    if (lane == 15) {
      #pragma unroll
      for (int v = 0; v < 8; ++v) zend[wv * 16 + v] = acc[v];
    } else if (lane == 31) {
      #pragma unroll
      for (int v = 0; v < 8; ++v) zend[wv * 16 + 8 + v] = acc[v];
    }
    __syncthreads();
    float ze[16];
    #pragma unroll
    for (int k = 0; k < 16; ++k) ze[k] = zend[wv * 16 + k];
    // Carry into segment k (wave-local): C[k] = a^16*C[k-1] + ze[k-1]
    float C[16];
    C[0] = carry_tile;
    #pragma unroll
    for (int k = 1; k < 16; ++k) C[k] = a16 * C[k - 1] + ze[k - 1];
    float Cu[8];
    #pragma unroll
    for (int v = 0; v < 8; ++v) Cu[v] = hi ? C[v + 8] : C[v];
    #pragma unroll
    for (int v = 0; v < 8; ++v) yreg[tau][v] = acc[v] + api * Cu[v];
    carry_tile = ze[15] + a16 * C[15];
  }

  if (lane == 0) Wend[wv] = carry_tile;
  __syncthreads();
  // Carry into this wave from earlier waves of the block (multiplier a^512)
  float wc = 0.0f;
  for (int v = 0; v < wv; ++v) wc = a512 * wc + Wend[v];

  // Emit block-local scan: w[t] = wave-local y[t] + a^(pos+1) * wc
  float* wrow = w_ws + (size_t)n * LEN + t0 + wv * WSAMP;
  #pragma unroll
  for (int tau = 0; tau < NTILE; ++tau) {
    float tf = (tau == 0) ? 1.0f : a256;
    #pragma unroll
    for (int v = 0; v < 8; ++v) {
      float fac = api * a16p[v] * tf;
      int pos = tau * TILE + (v + 8 * hi) * 16 + m;
      wrow[pos] = yreg[tau][v] + fac * wc;
    }
  }
  if (tid == 0) {
    float Eb = 0.0f;
    #pragma unroll
    for (int v = 0; v < WAVES; ++v) Eb = a512 * Eb + Wend[v];
    P_ws[n * NBLK + j] = Eb;   // block end value (block-local)
  }
}

// -------- Pass 2: inter-block carry recurrence (per row; 64 blocks) --------
__global__ void pass2_carry(const float* __restrict__ z_alpha,
                            const float* __restrict__ P_ws,
                            float* __restrict__ E_ws)
{
  int n = threadIdx.x;
  if (n >= N_BATCH) return;
  float a  = sigmoidf_(z_alpha[n]);
  float aC = exp2f((float)CHUNK * log2f(a));
  float E = 0.0f;
  for (int j = 0; j < NBLK; ++j) {
    E_ws[n * NBLK + j] = E;            // carry INTO block j
    E = aC * E + P_ws[n * NBLK + j];
  }
}

// -------- Pass 3: carry fixup + log-knee gain + apply to both channels --------
__global__ __launch_bounds__(256) void pass3_gain(
    const float* __restrict__ x, const float* __restrict__ z_alpha,
    const float* __restrict__ log_th, const float* __restrict__ log_ratio,
    const float* __restrict__ log_knee,
    const float* __restrict__ w_ws, const float* __restrict__ E_ws,
    float* __restrict__ out)
{
  const int per_row = LEN / 4;
  int gid = blockIdx.x * blockDim.x + threadIdx.x;
  int n = gid >> 16;                 // gid / (LEN/4), LEN/4 = 65536
  int t = (gid & (per_row - 1)) * 4;

  float a    = sigmoidf_(z_alpha[n]);
  float l2a  = log2f(a);
  float th   = log_th[n] - 6.0f;
  float invr = 1.0f / (1.0f + expf(log_ratio[n]) + 0.001f);
  float knee = expf(log_knee[n]);
  float kh   = 0.5f * knee;
  float im1  = invr - 1.0f;
  float qden = 1.0f / (2.0f * (knee + 0.001f));

  int   jb   = t >> 12;              // block index
  int   tl   = t & (CHUNK - 1);
  float Ec   = E_ws[n * NBLK + jb];
  float afac = exp2f((float)(tl + 1) * l2a);

  const float4 wv4 = *(const float4*)(w_ws + (size_t)n * LEN + t);
  const float* xb  = x + (size_t)n * 2 * LEN;
  float4 c0 = *(const float4*)(xb + t);
  float4 c1 = *(const float4*)(xb + LEN + t);

  float wsamp[4] = {wv4.x, wv4.y, wv4.z, wv4.w};
  float g[4];
  #pragma unroll
  for (int s = 0; s < 4; ++s) {
    float y  = wsamp[s] + afac * Ec;
    afac *= a;
    float le = logf(y + 1e-5f);
    float d  = le - th;
    float lg;
    if (d < -kh)      lg = 0.0f;
    else if (d > kh)  lg = d * im1;
    else { float u = d + kh; lg = im1 * u * u * qden; }
    g[s] = expf(lg);
  }
  float4 o0, o1;
  o0.x = g[0] * c0.x; o0.y = g[1] * c0.y; o0.z = g[2] * c0.z; o0.w = g[3] * c0.w;
  o1.x = g[0] * c1.x; o1.y = g[1] * c1.y; o1.z = g[2] * c1.z; o1.w = g[3] * c1.w;
  *(float4*)(out + (size_t)n * 2 * LEN + t)       = o0;
  *(float4*)(out + (size_t)n * 2 * LEN + LEN + t) = o1;
}

extern "C" void kernel_launch(void* const* d_in, const int* in_sizes, int n_in,
                              void* d_out, int out_size, void* d_ws, size_t ws_size,
                              hipStream_t stream)
{
  const float* x  = (const float*)d_in[0];
  const float* za = (const float*)d_in[1];
  const float* lt = (const float*)d_in[2];
  const float* lr = (const float*)d_in[3];
  const float* lk = (const float*)d_in[4];
  float* out = (float*)d_out;

  float* w_ws = (float*)d_ws;                        // N*LEN floats (32 MB)
  float* P_ws = w_ws + (size_t)N_BATCH * LEN;        // N*NBLK floats
  float* E_ws = P_ws + N_BATCH * NBLK;               // N*NBLK floats

  pass1_scan<<<dim3(NBLK, N_BATCH), 256, 0, stream>>>(x, za, w_ws, P_ws);
  pass2_carry<<<1, 32, 0, stream>>>(za, P_ws, E_ws);
  pass3_gain<<<(N_BATCH * (LEN / 4)) / 256, 256, 0, stream>>>(
      x, za, lt, lr, lk, w_ws, E_ws, out);
}